// HistoNetLoss_52501680226348
// MI455X (gfx1250) — compile-verified
//
#include <hip/hip_runtime.h>
#include <hip/hip_bf16.h>
#include <math.h>

// ---------------- problem constants (match reference) ----------------
#define TSIZE        151                  // NUM_STEPS
#define HALF_BINS    75                   // (NUM_STEPS-1)/2
#define SUPPORT      512                  // 64 classes * 8 support/class
#define TARGET       65024                // 65536 - 512
#define NUM_CLASSES  64
#define STEPF        (2.0f / 150.0f)      // 2/(NUM_STEPS-1), computed in f32 like XLA

typedef __attribute__((ext_vector_type(2))) float v2f;
typedef __attribute__((ext_vector_type(8))) float v8f;

// ---------------- histogram kernel ----------------
// grid: 508 blocks x 256 threads (8 wave32s). Each block owns 128 rows x 512 cols
// = 64K elements, streamed as float4 (global_load_b128, fully coalesced).
// Per-wave private LDS histograms (ds_add_f32) avoid inter-wave bank contention.
constexpr int HIST_ROWS    = 128;
constexpr int HIST_THREADS = 256;
constexpr int HIST_BLOCKS  = TARGET / HIST_ROWS;   // 508

__global__ __launch_bounds__(HIST_THREADS)
void hist_kernel(const float* __restrict__ ip, const int* __restrict__ labels,
                 float* __restrict__ g_pos, float* __restrict__ g_neg,
                 unsigned int* __restrict__ g_posCount)
{
    __shared__ float hist[2][8][160];      // [pos/neg][wave][bin(0..151 used)]
    __shared__ int   slab[SUPPORT];
    __shared__ int   tlab[HIST_ROWS];
    __shared__ unsigned int wred[8];

    const int tid  = threadIdx.x;
    const int wave = tid >> 5;
    float* hp = hist[0][wave];
    float* hn = hist[1][wave];

    for (int i = tid; i < 2 * 8 * 160; i += HIST_THREADS) (&hist[0][0][0])[i] = 0.0f;
    for (int i = tid; i < SUPPORT; i += HIST_THREADS) slab[i] = labels[i];
    const int rowBase = blockIdx.x * HIST_ROWS;
    for (int i = tid; i < HIST_ROWS; i += HIST_THREADS) tlab[i] = labels[SUPPORT + rowBase + i];
    __syncthreads();

    const float4* __restrict__ src = (const float4*)(ip + (size_t)rowBase * SUPPORT);
    const int NV4 = HIST_ROWS * SUPPORT / 4;   // 16384 float4 per block
    unsigned int posc = 0;

    for (int e4 = tid; e4 < NV4; e4 += HIST_THREADS) {
        float4 v = src[e4];
        int e  = e4 << 2;
        int r  = e >> 9;            // local row (cols = 512)
        int j  = e & 511;           // col of first element (multiple of 4)
        int tl = tlab[r];
        int4 sl4 = ((const int4*)slab)[j >> 2];
        float s[4]  = {v.x, v.y, v.z, v.w};
        int   sl[4] = {sl4.x, sl4.y, sl4.z, sl4.w};
#pragma unroll
        for (int q = 0; q < 4; ++q) {
            float t    = s[q] / STEPF;            // matches reference s/step
            float kf   = floorf(t);
            float frac = t - kf;
            int k = (int)kf + HALF_BINS;          // nominally in [75,149]
            k = min(max(k, 0), TSIZE - 1);        // safety clamp, k+1 <= 151
            bool  pos = (sl[q] == tl);
            float* h  = pos ? hp : hn;
            atomicAdd(&h[k],     frac);           // reference: bin k gets frac
            atomicAdd(&h[k + 1], 1.0f - frac);    // bin k+1 gets 1-frac
            posc += pos ? 1u : 0u;
        }
    }
    __syncthreads();

    // flush per-wave copies to global partial histograms (one atomic per bin per block)
    for (int b = tid; b < TSIZE + 1; b += HIST_THREADS) {
        float sp = 0.f, sn = 0.f;
#pragma unroll
        for (int w = 0; w < 8; ++w) { sp += hist[0][w][b]; sn += hist[1][w][b]; }
        atomicAdd(&g_pos[b], sp);
        atomicAdd(&g_neg[b], sn);
    }

    // exact integer pos count
    unsigned int pv = posc;
    for (int off = 16; off; off >>= 1) pv += __shfl_down(pv, off);
    if ((tid & 31) == 0) wred[wave] = pv;
    __syncthreads();
    if (tid == 0) {
        unsigned int t = 0;
        for (int w = 0; w < 8; ++w) t += wred[w];
        atomicAdd(g_posCount, t);
    }
}

// ---------------- direct (cross-entropy) loss kernel ----------------
// one thread per row: 16 x float4 loads (64 logits), max -> logsumexp -> nll.
constexpr int NLL_THREADS = 256;
constexpr int NLL_BLOCKS  = TARGET / NLL_THREADS;  // 254

__global__ __launch_bounds__(NLL_THREADS)
void nll_kernel(const float* __restrict__ logits, const int* __restrict__ labels,
                float* __restrict__ g_nll)
{
    __shared__ float sred[8];
    const int row = blockIdx.x * NLL_THREADS + threadIdx.x;
    float nll = 0.f;
    if (row < TARGET) {
        const float4* __restrict__ p = (const float4*)(logits + (size_t)row * NUM_CLASSES);
        float4 v[16];
        float mx = -INFINITY;
#pragma unroll
        for (int i = 0; i < 16; ++i) {
            v[i] = p[i];
            mx = fmaxf(mx, fmaxf(fmaxf(v[i].x, v[i].y), fmaxf(v[i].z, v[i].w)));
        }
        float se = 0.f;
#pragma unroll
        for (int i = 0; i < 16; ++i) {
            se += expf(v[i].x - mx) + expf(v[i].y - mx)
                + expf(v[i].z - mx) + expf(v[i].w - mx);
        }
        float lse = mx + logf(se);
        int lab = labels[SUPPORT + row];
        float xv = logits[(size_t)row * NUM_CLASSES + lab];   // gather via L2, avoids scratch spill
        nll = lse - xv;
    }
    for (int off = 16; off; off >>= 1) nll += __shfl_down(nll, off);
    if ((threadIdx.x & 31) == 0) sred[threadIdx.x >> 5] = nll;
    __syncthreads();
    if (threadIdx.x == 0) {
        float t = 0.f;
        for (int w = 0; w < 8; ++w) t += sred[w];
        atomicAdd(g_nll, t);
    }
}

// ---------------- final kernel: WMMA triangular cumsum + dot ----------------
// Single wave32 (EXEC all ones, required by WMMA). cdf over 151 bins is done as
// 10 blocks of a 16x16 lower-triangular mat-vec via chained V_WMMA_F32_16X16X4_F32:
//   A[m][k] = hp_block[k]  (replicated over rows)   -> a VGPR0: K0/K2, VGPR1: K1/K3
//   B[k][n] = (k <= n)                              -> triangular ones mask
//   D[m][n] = local_cdf[n] for every m; c[0] at lane n gives cdf[n].
// Off-diagonal all-ones tiles collapse to a scalar block-prefix (spref).
__global__ __launch_bounds__(32)
void final_kernel(const float* __restrict__ g_pos, const float* __restrict__ g_neg,
                  const unsigned int* __restrict__ g_posCount,
                  const float* __restrict__ g_nll, float* __restrict__ out)
{
    __shared__ float shp[160];
    __shared__ float shn[160];
    __shared__ float spref[10];

    const int lane = threadIdx.x;
    const float posNum = (float)(*g_posCount);
    const float totalf = (float)((long long)TARGET * (long long)SUPPORT);
    const float negNum = totalf - posNum;

    for (int i = lane; i < 160; i += 32) {
        shp[i] = (i < TSIZE) ? g_pos[i] / posNum : 0.0f;
        shn[i] = (i < TSIZE) ? g_neg[i] / negNum : 0.0f;
    }
    __syncthreads();
    if (lane == 0) {
        float acc = 0.f;
        for (int jb = 0; jb < 10; ++jb) {
            spref[jb] = acc;
            for (int q = 0; q < 16; ++q) acc += shp[jb * 16 + q];
        }
    }
    __syncthreads();

    const int n     = lane & 15;     // column within 16-bin block
    const int khalf = lane >> 4;     // 0: K{0,1}, 1: K{2,3} per f32 A-matrix layout
    float result = 0.f;

    for (int jb = 0; jb < 10; ++jb) {
        v8f c = {};
#pragma unroll
        for (int kb = 0; kb < 4; ++kb) {
            const int k0 = kb * 4 + khalf * 2;
            v2f a, b;
            a.x = shp[jb * 16 + k0];
            a.y = shp[jb * 16 + k0 + 1];
            b.x = (k0     <= n) ? 1.0f : 0.0f;   // lower-triangular (incl. diagonal)
            b.y = (k0 + 1 <= n) ? 1.0f : 0.0f;
            c = __builtin_amdgcn_wmma_f32_16x16x4_f32(
                    /*neg_a=*/false, a, /*neg_b=*/false, b,
                    /*c_mod=*/(short)0, c, /*reuse_a=*/false, /*reuse_b=*/false);
        }
        const float cdfv = c[0];                 // lane n holds local cdf[n]
        const int gbin = jb * 16 + n;
        float contrib = (lane < 16 && gbin < TSIZE) ? (spref[jb] + cdfv) * shn[gbin] : 0.f;
        result += contrib;                       // lanes 16-31 masked (duplicates)
    }
    for (int off = 16; off; off >>= 1) result += __shfl_down(result, off);
    if (lane == 0) {
        out[0] = result;                                   // histo_loss
        out[1] = g_nll[0] * (1.0f / (float)TARGET);        // direct_loss (weight 1.0)
    }
}

// ---------------- launch ----------------
extern "C" void kernel_launch(void* const* d_in, const int* in_sizes, int n_in,
                              void* d_out, int out_size, void* d_ws, size_t ws_size,
                              hipStream_t stream)
{
    const float* logits = (const float*)d_in[0];   // (TARGET, 64) f32
    const float* inner  = (const float*)d_in[1];   // (TARGET, 512) f32
    const int*   labels = (const int*)d_in[2];     // (65536,) int32
    float* out = (float*)d_out;                    // [histo_loss, direct_loss]

    float*        g_pos      = (float*)d_ws;                   // 160 f32
    float*        g_neg      = g_pos + 160;                    // 160 f32
    float*        g_nll      = g_neg + 160;                    // 1 f32
    unsigned int* g_posCount = (unsigned int*)(g_nll + 1);     // 1 u32

    hipMemsetAsync(d_ws, 0, (160 + 160 + 2) * sizeof(float), stream);

    hist_kernel<<<HIST_BLOCKS, HIST_THREADS, 0, stream>>>(inner, labels, g_pos, g_neg, g_posCount);
    nll_kernel<<<NLL_BLOCKS, NLL_THREADS, 0, stream>>>(logits, labels, g_nll);
    final_kernel<<<1, 32, 0, stream>>>(g_pos, g_neg, g_posCount, g_nll, out);
}